// PathwayNegativeSamplingLoss_61727269978175
// MI455X (gfx1250) — compile-verified
//
#include <hip/hip_runtime.h>

typedef __attribute__((ext_vector_type(2))) float v2f;
typedef __attribute__((ext_vector_type(8))) float v8f;

#define DIM   128
#define KSTEP 4
#define NNEG  10
#define NDOT  (NNEG + 1)
#define WAVES_PER_BLOCK 8
#define TPB   (WAVES_PER_BLOCK * 32)

__device__ __forceinline__ float log_sigmoid(float x) {
    // stable: min(x,0) - log1p(exp(-|x|))
    return fminf(x, 0.0f) - log1pf(__expf(-fabsf(x)));
}

__device__ __forceinline__ float diag_sel(v8f c, int r) {
    float d = c[0];
    d = (r == 1) ? c[1] : d;
    d = (r == 2) ? c[2] : d;
    d = (r == 3) ? c[3] : d;
    d = (r == 4) ? c[4] : d;
    d = (r == 5) ? c[5] : d;
    d = (r == 6) ? c[6] : d;
    d = (r == 7) ? c[7] : d;
    return d;
}

// One wave handles 16 samples of one term: 11 accumulated 16x16 WMMA chains
// over K=128 (step 4); the diagonal of each 16x16 result is the per-sample dot.
__global__ __launch_bounds__(TPB) void sgns_wmma_kernel(
    const float* __restrict__ gene_e, const float* __restrict__ path_e,
    const float* __restrict__ gene_w, const float* __restrict__ path_w,
    const int* __restrict__ gg_src, const int* __restrict__ gg_ctx,
    const int* __restrict__ gp_g,  const int* __restrict__ gp_p,
    const int* __restrict__ pg_p,  const int* __restrict__ pg_g,
    const int* __restrict__ neg_gg, const int* __restrict__ neg_gp,
    const int* __restrict__ neg_pg,
    float* __restrict__ partials, int B, int groups)
{
    const int term = blockIdx.y;
    const float* hE; const float* wE; const int* hI; const int* wI; const int* nI;
    float weight;
    if (term == 0)      { hE = gene_e; wE = gene_w; hI = gg_src; wI = gg_ctx; nI = neg_gg; weight = 1.0f; }
    else if (term == 1) { hE = gene_e; wE = path_w; hI = gp_g;  wI = gp_p;  nI = neg_gp; weight = 1.0f; }
    else                { hE = path_e; wE = gene_w; hI = pg_p;  wI = pg_g;  nI = neg_pg; weight = 0.5f; }

    const int lane = threadIdx.x & 31;
    const int wv   = threadIdx.x >> 5;
    const int mn   = lane & 15;            // A row (M) / B col (N) for this lane
    const int koff = (lane >> 4) * 2;      // K offset inside each 4-wide chunk

    // Diagonal ownership: lanes 0..7 hold diag samples 0..7 (vgpr=lane),
    // lanes 24..31 hold diag samples 8..15 (vgpr=lane-24).
    const int  drow   = (lane < 8) ? lane : (lane - 16);
    const bool dvalid = (lane < 8) || (lane >= 24);
    const int  dreg   = (lane < 8) ? lane : (lane - 24);

    float wavesum = 0.0f;

    for (int g = blockIdx.x * WAVES_PER_BLOCK + wv; g < groups;
         g += gridDim.x * WAVES_PER_BLOCK) {

        int sA = g * 16 + mn;
        if (sA >= B) sA = B - 1;   // clamp tail; masked at accumulation

        // 32-bit element offsets (tables are << 4G elements); kept in registers
        // after unrolling, so address-space inference sees direct GEPs off the
        // kernel-arg bases -> global_load_b64 (not flat), saddr-friendly.
        const unsigned offA = (unsigned)hI[sA] * DIM + (unsigned)koff;
        unsigned offB[NDOT];
        offB[0] = (unsigned)wI[sA] * DIM + (unsigned)koff;
#pragma unroll
        for (int j = 0; j < NNEG; ++j)
            offB[1 + j] = (unsigned)nI[sA * NNEG + j] * DIM + (unsigned)koff;

        v8f acc[NDOT] = {};

#pragma unroll 2
        for (int k0 = 0; k0 < DIM; k0 += KSTEP) {
            const float* hk = hE + k0;   // uniform per-iteration base
            const float* wk = wE + k0;

            // ---- load phase: 12 independent b64 loads in flight ----
            v2f a = *(const v2f*)(hk + offA);
            v2f b[NDOT];
#pragma unroll
            for (int j = 0; j < NDOT; ++j)
                b[j] = *(const v2f*)(wk + offB[j]);

            // ---- compute phase: 11 WMMAs, one wait before the group ----
#pragma unroll
            for (int j = 0; j < NDOT; ++j)
                acc[j] = __builtin_amdgcn_wmma_f32_16x16x4_f32(
                    false, a, false, b[j], (short)0, acc[j], false, false);
        }

        float ls = log_sigmoid(diag_sel(acc[0], dreg));   // positive
#pragma unroll
        for (int j = 1; j < NDOT; ++j)
            ls += log_sigmoid(-diag_sel(acc[j], dreg));   // negatives

        const int samp = g * 16 + drow;
        wavesum += (dvalid && samp < B) ? ls : 0.0f;
    }

    // wave32 reduce
#pragma unroll
    for (int off = 16; off >= 1; off >>= 1)
        wavesum += __shfl_xor(wavesum, off, 32);

    __shared__ float smem[WAVES_PER_BLOCK];
    if (lane == 0) smem[wv] = wavesum;
    __syncthreads();
    if (threadIdx.x == 0) {
        float t = 0.0f;
#pragma unroll
        for (int i = 0; i < WAVES_PER_BLOCK; ++i) t += smem[i];
        partials[(long long)blockIdx.y * gridDim.x + blockIdx.x] =
            -weight * t / (float)B;
    }
}

// Deterministic single-block final reduction (no atomics on the scalar output).
__global__ __launch_bounds__(256) void sgns_reduce_kernel(
    const float* __restrict__ partials, int n, float* __restrict__ out)
{
    __shared__ float smem[256];
    float t = 0.0f;
    for (int i = threadIdx.x; i < n; i += 256) t += partials[i];
    smem[threadIdx.x] = t;
    __syncthreads();
    for (int s = 128; s > 0; s >>= 1) {
        if ((int)threadIdx.x < s) smem[threadIdx.x] += smem[threadIdx.x + s];
        __syncthreads();
    }
    if (threadIdx.x == 0) out[0] = smem[0];
}

extern "C" void kernel_launch(void* const* d_in, const int* in_sizes, int n_in,
                              void* d_out, int out_size, void* d_ws, size_t ws_size,
                              hipStream_t stream) {
    const float* gene_e = (const float*)d_in[0];
    const float* path_e = (const float*)d_in[1];
    const float* gene_w = (const float*)d_in[2];
    const float* path_w = (const float*)d_in[3];
    const int* gg_src = (const int*)d_in[4];
    const int* gg_ctx = (const int*)d_in[5];
    const int* gp_g   = (const int*)d_in[6];
    const int* gp_p   = (const int*)d_in[7];
    const int* pg_p   = (const int*)d_in[8];
    const int* pg_g   = (const int*)d_in[9];
    const int* neg_gg = (const int*)d_in[10];
    const int* neg_gp = (const int*)d_in[11];
    const int* neg_pg = (const int*)d_in[12];

    const int B = in_sizes[4];
    const int groups = (B + 15) / 16;
    const int gx = (groups + WAVES_PER_BLOCK - 1) / WAVES_PER_BLOCK;

    float* partials = (float*)d_ws;

    dim3 grid(gx, 3);
    sgns_wmma_kernel<<<grid, TPB, 0, stream>>>(
        gene_e, path_e, gene_w, path_w,
        gg_src, gg_ctx, gp_g, gp_p, pg_p, pg_g,
        neg_gg, neg_gp, neg_pg,
        partials, B, groups);

    sgns_reduce_kernel<<<1, 256, 0, stream>>>(partials, gx * 3, (float*)d_out);
}